// MultiHeadSelfAttention_62045097558208
// MI455X (gfx1250) — compile-verified
//
#include <hip/hip_runtime.h>
#include <math.h>
#include <stdint.h>

typedef __attribute__((ext_vector_type(2))) float v2f;
typedef __attribute__((ext_vector_type(4))) float v4f;
typedef __attribute__((ext_vector_type(8))) float v8f;

#define B_  4
#define L_  2048
#define H_  1024
#define NH_ 16
#define HD_ 64
#define M_  (B_ * L_)   // 8192
#define K_  H_          // 1024

static __device__ __forceinline__ float neg_inf() { return -__builtin_inff(); }

// ---------------------------------------------------------------------------
// 16-byte global -> LDS copy. Async path FORCED on the device pass this round:
// if the builtin is absent/mismatched we want a loud compile error, not a
// silent fallback. Host pass keeps the plain copy (it never emits device code).
// ---------------------------------------------------------------------------
#if defined(__HIP_DEVICE_COMPILE__)
#define USE_ASYNC_LDS 1
#pragma message("CDNA5 device pass: global_load_async_to_lds_b128 FORCED ON")
typedef int v4i __attribute__((ext_vector_type(4)));
typedef __attribute__((address_space(1))) v4i* gv4_t;
typedef __attribute__((address_space(3))) v4i* lv4_t;
static __device__ __forceinline__ void cp16(void* lds_dst, const void* gsrc) {
  __builtin_amdgcn_global_load_async_to_lds_b128(
      (gv4_t)(uintptr_t)gsrc, (lv4_t)(uint32_t)(uintptr_t)lds_dst, 0, 0);
}
static __device__ __forceinline__ void cp_wait() {
  __builtin_amdgcn_s_wait_asynccnt(0);
}
#else
#define USE_ASYNC_LDS 0
static __device__ __forceinline__ void cp16(void* lds_dst, const void* gsrc) {
  *(v4f*)lds_dst = *(const v4f*)gsrc;
}
static __device__ __forceinline__ void cp_wait() {}
#endif

// ---------------------------------------------------------------------------
// GEMM: out = X @ W^T + bias
//   X: (M_ x K_) row-major, W: (H_ x K_) row-major, bias: (H_)
//   MODE 0: scatter result to (B, NH, L, HD);  MODE 1: row-major (M_ x H_)
// 256 threads = 8 waves in a 4x2 grid; block tile 128(M) x 128(N); each wave
// owns a 32x64 region = 2x4 WMMA accumulators. K streamed in double-buffered
// 32-wide LDS chunks (rows padded to 36 floats: 16B-aligned, conflict-free).
// ---------------------------------------------------------------------------
template <int MODE>
__global__ __launch_bounds__(256) void gemm_xwt(const float* __restrict__ X,
                                                const float* __restrict__ W,
                                                const float* __restrict__ bias,
                                                float* __restrict__ out) {
  __shared__ float Xs[2][128][36];
  __shared__ float Ws[2][128][36];

  const int tid   = threadIdx.x;
  const int wid   = tid >> 5;
  const int lane  = tid & 31;
  const int lidx  = lane & 15;
  const int khalf = lane >> 4;
  const int qw    = wid >> 1;   // 0..3 : m 32-row slice
  const int nw    = wid & 1;    // 0..1 : n 64-col slice
  const int m0    = blockIdx.y * 128;
  const int n0    = blockIdx.x * 128;

  v8f acc[2][4] = {};

  auto stage = [&](int buf, int k0) {
    for (int i = tid; i < 2048; i += 256) {   // 1024 v4 for X, 1024 v4 for W
      const int r = (i & 1023) >> 3, c = (i & 7) * 4;
      if (i < 1024) cp16(&Xs[buf][r][c], &X[(size_t)(m0 + r) * K_ + k0 + c]);
      else          cp16(&Ws[buf][r][c], &W[(size_t)(n0 + r) * K_ + k0 + c]);
    }
  };

  stage(0, 0);
  cp_wait();
  __syncthreads();

  for (int k0 = 0; k0 < K_; k0 += 32) {
    const int cur = (k0 >> 5) & 1;
    if (k0 + 32 < K_) stage(cur ^ 1, k0 + 32);

#pragma unroll
    for (int k4 = 0; k4 < 8; ++k4) {
      const int kk = k4 * 4 + 2 * khalf;
      v2f a[2], b[4];
#pragma unroll
      for (int t = 0; t < 2; ++t)
        a[t] = *(const v2f*)&Xs[cur][(2 * qw + t) * 16 + lidx][kk];
#pragma unroll
      for (int t = 0; t < 4; ++t)
        b[t] = *(const v2f*)&Ws[cur][(4 * nw + t) * 16 + lidx][kk];
#pragma unroll
      for (int ti = 0; ti < 2; ++ti)
#pragma unroll
        for (int tj = 0; tj < 4; ++tj)
          acc[ti][tj] = __builtin_amdgcn_wmma_f32_16x16x4_f32(
              false, a[ti], false, b[tj], (short)0, acc[ti][tj], false, false);
    }
    cp_wait();
    __syncthreads();
  }

#pragma unroll
  for (int ti = 0; ti < 2; ++ti)
#pragma unroll
    for (int tj = 0; tj < 4; ++tj)
#pragma unroll
      for (int r = 0; r < 8; ++r) {
        const int m = m0 + (2 * qw + ti) * 16 + r + 8 * khalf;
        const int n = n0 + (4 * nw + tj) * 16 + lidx;
        const float v = acc[ti][tj][r] + bias[n];
        if (MODE == 0) {
          const int bi = m / L_, li = m % L_;
          const int h = n / HD_, d = n % HD_;
          out[((size_t)(bi * NH_ + h) * L_ + li) * HD_ + d] = v;
        } else {
          out[(size_t)m * H_ + n] = v;
        }
      }
}

// ---------------------------------------------------------------------------
// Flash attention: one block per (b, h, 128-query tile); 64-key chunks of
// K/V/mask double-buffered through LDS (async). 8 waves in a 4x2 grid; each
// wave owns 32 q-rows x 32 cols (2x2 WMMA tiles) for S and O. Online softmax
// (2 threads per row, hardware __expf) with running row max/sum; fully-masked
// rows -> output 0. mask is jax bool -> 1 byte per element, shape (B,1,L,L).
// ---------------------------------------------------------------------------
__global__ __launch_bounds__(256) void attention(const float* __restrict__ Q,
                                                 const float* __restrict__ Kb,
                                                 const float* __restrict__ Vb,
                                                 const unsigned char* __restrict__ mask,
                                                 float* __restrict__ out) {
  __shared__ float Qs[128][68];    // rows 272B: 16B-aligned, conflict-free
  __shared__ float Ks[2][64][68];
  __shared__ float Vs[2][64][68];
  __shared__ float Ss[128][68];
  __shared__ unsigned char Ms[2][128][64];
  __shared__ float m_run[128], l_run[128], corr[128];
  __shared__ float pmax[128][2], psum[128][2];

  const int tid   = threadIdx.x;
  const int wid   = tid >> 5;
  const int lane  = tid & 31;
  const int lidx  = lane & 15;
  const int khalf = lane >> 4;
  const int qw    = wid >> 1;   // 0..3 : 32-query slice
  const int cw    = wid & 1;    // 0..1 : 32-col slice (keys for S, dims for O)
  const int q0    = blockIdx.x * 128;
  const int hh    = blockIdx.y;
  const int bb    = blockIdx.z;

  const float* Qp = Q + (size_t)(bb * NH_ + hh) * L_ * HD_;
  const float* Kp = Kb + (size_t)(bb * NH_ + hh) * L_ * HD_;
  const float* Vp = Vb + (size_t)(bb * NH_ + hh) * L_ * HD_;
  const unsigned char* Mp = mask + (size_t)bb * L_ * L_;

  auto stage_kvm = [&](int buf, int kb) {
    for (int i = tid; i < 2560; i += 256) {   // 1024 v4 K + 1024 v4 V + 512 v4 mask
      if (i < 1024) {
        const int r = i >> 4, c = (i & 15) * 4;
        cp16(&Ks[buf][r][c], &Kp[(size_t)(kb + r) * HD_ + c]);
      } else if (i < 2048) {
        const int j = i - 1024;
        const int r = j >> 4, c = (j & 15) * 4;
        cp16(&Vs[buf][r][c], &Vp[(size_t)(kb + r) * HD_ + c]);
      } else {
        const int j = i - 2048;
        const int r = j >> 2, c = (j & 3) * 16;        // bytes
        cp16(&Ms[buf][r][c], &Mp[(size_t)(q0 + r) * L_ + kb + c]);
      }
    }
  };

  for (int i = tid; i < 2048; i += 256) {     // Q tile: 128 x 16 v4
    const int r = i >> 4, c = (i & 15) * 4;
    cp16(&Qs[r][c], &Qp[(size_t)(q0 + r) * HD_ + c]);
  }
  stage_kvm(0, 0);
  if (tid < 128) { m_run[tid] = neg_inf(); l_run[tid] = 0.f; }
  cp_wait();
  __syncthreads();

  v8f o[2][2] = {};

  for (int kb = 0; kb < L_; kb += 64) {
    const int cur = (kb >> 6) & 1;
    if (kb + 64 < L_) stage_kvm(cur ^ 1, kb + 64);

    // ---- S = (Q K^T) / sqrt(HD), masked to -inf --------------------------
    v8f s[2][2] = {};
#pragma unroll
    for (int d4 = 0; d4 < 16; ++d4) {
      const int kk = d4 * 4 + 2 * khalf;
      v2f a[2], b[2];
#pragma unroll
      for (int t = 0; t < 2; ++t)
        a[t] = *(const v2f*)&Qs[(2 * qw + t) * 16 + lidx][kk];
#pragma unroll
      for (int t = 0; t < 2; ++t)
        b[t] = *(const v2f*)&Ks[cur][(2 * cw + t) * 16 + lidx][kk];
#pragma unroll
      for (int ti = 0; ti < 2; ++ti)
#pragma unroll
        for (int tj = 0; tj < 2; ++tj)
          s[ti][tj] = __builtin_amdgcn_wmma_f32_16x16x4_f32(
              false, a[ti], false, b[tj], (short)0, s[ti][tj], false, false);
    }
#pragma unroll
    for (int ti = 0; ti < 2; ++ti)
#pragma unroll
      for (int tj = 0; tj < 2; ++tj)
#pragma unroll
        for (int r = 0; r < 8; ++r) {
          const int qr = (2 * qw + ti) * 16 + r + 8 * khalf;
          const int kc = (2 * cw + tj) * 16 + lidx;
          Ss[qr][kc] = Ms[cur][qr][kc] ? s[ti][tj][r] * 0.125f : neg_inf();
        }
    __syncthreads();

    // ---- Online softmax update: 2 threads per row, 32 cols each ----------
    {
      const int srow = tid >> 1, shalf = (tid & 1) * 32;
      float mx = neg_inf();
      for (int j = 0; j < 32; ++j) mx = fmaxf(mx, Ss[srow][shalf + j]);
      pmax[srow][tid & 1] = mx;
      __syncthreads();

      const float m_new = fmaxf(m_run[srow], fmaxf(pmax[srow][0], pmax[srow][1]));
      float sum = 0.f;
      if (m_new == neg_inf()) {
        for (int j = 0; j < 32; ++j) Ss[srow][shalf + j] = 0.f;
      } else {
        for (int j = 0; j < 32; ++j) {
          const float p = __expf(Ss[srow][shalf + j] - m_new);  // v_exp_f32
          Ss[srow][shalf + j] = p;
          sum += p;
        }
      }
      psum[srow][tid & 1] = sum;
      __syncthreads();

      if ((tid & 1) == 0) {
        const float c = (m_new == neg_inf()) ? 1.f : __expf(m_run[srow] - m_new);
        l_run[srow] = l_run[srow] * c + psum[srow][0] + psum[srow][1];
        m_run[srow] = m_new;
        corr[srow] = c;
      }
    }
    __syncthreads();

    // ---- Rescale O, then O += P @ V --------------------------------------
#pragma unroll
    for (int ti = 0; ti < 2; ++ti)
#pragma unroll
      for (int tj = 0; tj < 2; ++tj)
#pragma unroll
        for (int r = 0; r < 8; ++r)
          o[ti][tj][r] *= corr[(2 * qw + ti) * 16 + r + 8 * khalf];

#pragma unroll
    for (int k4 = 0; k4 < 16; ++k4) {
      const int kk = k4 * 4 + 2 * khalf;
      v2f a[2], b[2];
#pragma unroll
      for (int t = 0; t < 2; ++t)
        a[t] = *(const v2f*)&Ss[(2 * qw + t) * 16 + lidx][kk];
#pragma unroll
      for (int t = 0; t < 2; ++t) {
        b[t].x = Vs[cur][kk + 0][(2 * cw + t) * 16 + lidx];
        b[t].y = Vs[cur][kk + 1][(2 * cw + t) * 16 + lidx];
      }
#pragma unroll
      for (int ti = 0; ti < 2; ++ti)
#pragma unroll
        for (int tj = 0; tj < 2; ++tj)
          o[ti][tj] = __builtin_amdgcn_wmma_f32_16x16x4_f32(
              false, a[ti], false, b[tj], (short)0, o[ti][tj], false, false);
    }
    cp_wait();
    __syncthreads();
  }

  // ---- Per-row reciprocal once, then normalize + write (B, L, H) ---------
  if (tid < 128) {
    const float l = l_run[tid];
    corr[tid] = (l > 0.f) ? 1.f / l : 0.f;   // reuse corr[] as 1/l
  }
  __syncthreads();

#pragma unroll
  for (int ti = 0; ti < 2; ++ti)
#pragma unroll
    for (int tj = 0; tj < 2; ++tj)
#pragma unroll
      for (int r = 0; r < 8; ++r) {
        const int qr = (2 * qw + ti) * 16 + r + 8 * khalf;
        const float v = o[ti][tj][r] * corr[qr];
        const int d = (2 * cw + tj) * 16 + lidx;
        out[((size_t)bb * L_ + (q0 + qr)) * H_ + hh * HD_ + d] = v;
      }
}

// ---------------------------------------------------------------------------
extern "C" void kernel_launch(void* const* d_in, const int* in_sizes, int n_in,
                              void* d_out, int out_size, void* d_ws, size_t ws_size,
                              hipStream_t stream) {
  const float* x  = (const float*)d_in[0];
  const unsigned char* mask = (const unsigned char*)d_in[1];  // jax bool = 1 byte
  const float* Wq = (const float*)d_in[2];
  const float* bq = (const float*)d_in[3];
  const float* Wk = (const float*)d_in[4];
  const float* bk = (const float*)d_in[5];
  const float* Wv = (const float*)d_in[6];
  const float* bv = (const float*)d_in[7];
  const float* Wo = (const float*)d_in[8];
  const float* bo = (const float*)d_in[9];
  float* out = (float*)d_out;

  const size_t per = (size_t)B_ * L_ * H_;  // 8.4M floats = 32 MB
  float* qbuf = (float*)d_ws;
  float* kbuf = qbuf + per;
  float* vbuf = kbuf + per;
  float* abuf = vbuf + per;

  const dim3 gg(H_ / 128, M_ / 128);  // 8 x 64 blocks
  gemm_xwt<0><<<gg, 256, 0, stream>>>(x, Wq, bq, qbuf);
  gemm_xwt<0><<<gg, 256, 0, stream>>>(x, Wk, bk, kbuf);
  gemm_xwt<0><<<gg, 256, 0, stream>>>(x, Wv, bv, vbuf);

  attention<<<dim3(L_ / 128, NH_, B_), 256, 0, stream>>>(qbuf, kbuf, vbuf, mask, abuf);

  gemm_xwt<1><<<gg, 256, 0, stream>>>(abuf, Wo, bo, out);
}